// Model_66048007078267
// MI455X (gfx1250) — compile-verified
//
#include <hip/hip_runtime.h>
#include <math.h>

typedef __attribute__((ext_vector_type(2))) float v2f;
typedef __attribute__((ext_vector_type(8))) float v8f;

enum { NU_N = 100000, NM_N = 20000, E_N = 250000, EL_N = 200000 };

static inline int ceil_div_i(int a, int b) { return (a + b - 1) / b; }

// ---------------- CDNA5 async global->LDS helpers (inline asm) ----------------
// LDS address is the byte offset held in the VDST vgpr; generic shared-ptr low 32 bits == LDS offset.
__device__ inline void async_ld_b128(unsigned lds_byte_off, const float* gptr) {
  asm volatile("global_load_async_to_lds_b128 %0, %1, off"
               :: "v"(lds_byte_off), "v"(gptr)
               : "memory");
}
__device__ inline void wait_async0() {
  asm volatile("s_wait_asynccnt 0x0" ::: "memory");
}

// ---------------------------------------------------------------- fill
__global__ void k_fill(float* __restrict__ p, float v, int n) {
  int i = blockIdx.x * blockDim.x + threadIdx.x;
  if (i < n) p[i] = v;
}

// ------------------------------------------- fp32 WMMA GEMM (16x16x4), direct
// C[M,N] = A[M,K] @ B[K,N] (+bias[N]) (+relu).  M%16==0, N%16==0, K%4==0.
__global__ void k_gemm_f32(const float* __restrict__ A, const float* __restrict__ B,
                           const float* __restrict__ bias, float* __restrict__ C,
                           int M, int K, int N, int act) {
  int wave   = blockIdx.x * (blockDim.x >> 5) + (threadIdx.x >> 5);
  int tilesN = N >> 4;
  int tiles  = (M >> 4) * tilesN;
  if (wave >= tiles) return;                // wave-uniform: EXEC stays all-ones
  int mt = wave / tilesN, nt = wave - mt * tilesN;
  int lane = threadIdx.x & 31;
  int hf   = lane >> 4;                     // 0 -> K 0,1 ; 1 -> K 2,3
  int l15  = lane & 15;
  size_t arow = (size_t)(mt * 16 + l15) * (size_t)K;
  int ncol = nt * 16 + l15;
  v8f acc = {};
  for (int k = 0; k < K; k += 4) {
    int kb = k + hf * 2;
    v2f a, b;
    a[0] = A[arow + kb];
    a[1] = A[arow + kb + 1];
    b[0] = B[(size_t)kb * N + ncol];
    b[1] = B[(size_t)(kb + 1) * N + ncol];
    acc = __builtin_amdgcn_wmma_f32_16x16x4_f32(false, a, false, b, (short)0, acc,
                                                false, false);
  }
  float bv = bias ? bias[ncol] : 0.f;
  int row0 = mt * 16 + hf * 8;
#pragma unroll
  for (int r = 0; r < 8; ++r) {
    float v = acc[r] + bv;
    if (act) v = v > 0.f ? v : 0.f;
    C[(size_t)(row0 + r) * N + ncol] = v;
  }
}

// ---------------- fp32 WMMA GEMM with async-LDS-staged A panel ----------------
// Requires: M%16==0, N%64==0, K%32==0. Block = 128 threads (4 waves).
// Block computes a 16x64 tile; the shared 16xKT A panel is staged into LDS via
// global_load_async_to_lds_b128 and consumed by all 4 waves (ds_load).
#define KT 32
#define APAD 36  // padded row stride (floats): 144B, 16B aligned, bank-conflict free
__global__ void k_gemm_f32_lds(const float* __restrict__ A, const float* __restrict__ B,
                               const float* __restrict__ bias, float* __restrict__ C,
                               int M, int K, int N, int act) {
  __shared__ float tA[16 * APAD];
  int tilesN4 = N >> 6;
  int mt = blockIdx.x / tilesN4;
  int ng = blockIdx.x - mt * tilesN4;
  int wid  = threadIdx.x >> 5;
  int lane = threadIdx.x & 31;
  int hf   = lane >> 4;
  int l15  = lane & 15;
  int nt   = ng * 4 + wid;
  int ncol = nt * 16 + l15;
  // A-panel staging assignment: 128 threads x float4 = 16 rows x 32 cols
  int arow_ld = threadIdx.x >> 3;        // 0..15
  int acol_ld = (threadIdx.x & 7) << 2;  // 0,4,...,28
  const float* Abase = A + (size_t)(mt * 16 + arow_ld) * (size_t)K + acol_ld;
  unsigned ldsoff = (unsigned)(uintptr_t)(&tA[arow_ld * APAD + acol_ld]);
  v8f acc = {};
  for (int k0 = 0; k0 < K; k0 += KT) {
    async_ld_b128(ldsoff, Abase + k0);
    wait_async0();
    __syncthreads();
#pragma unroll
    for (int kt = 0; kt < KT; kt += 4) {
      int kb = kt + hf * 2;
      v2f a, b;
      a[0] = tA[l15 * APAD + kb];
      a[1] = tA[l15 * APAD + kb + 1];
      size_t bk = (size_t)(k0 + kb) * N + ncol;
      b[0] = B[bk];
      b[1] = B[bk + N];
      acc = __builtin_amdgcn_wmma_f32_16x16x4_f32(false, a, false, b, (short)0, acc,
                                                  false, false);
    }
    __syncthreads();
  }
  float bv = bias ? bias[ncol] : 0.f;
  int row0 = mt * 16 + hf * 8;
#pragma unroll
  for (int r = 0; r < 8; ++r) {
    float v = acc[r] + bv;
    if (act) v = v > 0.f ? v : 0.f;
    C[(size_t)(row0 + r) * N + ncol] = v;
  }
}

// ---------------------------------- per-(node,head) attention logit dot
__global__ void k_rowdot(const float* __restrict__ h, const float* __restrict__ a,
                         float* __restrict__ o, int n, int H, int C) {
  int i = blockIdx.x * blockDim.x + threadIdx.x;
  if (i >= n * H) return;
  int node = i / H, hd = i - node * H;
  const float* hp = h + (size_t)node * H * C + (size_t)hd * C;
  const float* ap = a + (size_t)hd * C;
  float acc = 0.f;
  for (int c = 0; c < C; ++c) acc += hp[c] * ap[c];
  o[i] = acc;
}

// ---------------------------------------------- float atomic max helper
__device__ inline void atomicMaxF(float* addr, float v) {
  if (v >= 0.f)
    atomicMax((int*)addr, __float_as_int(v));
  else
    atomicMin((unsigned int*)addr, __float_as_uint(v));
}

// -------------------------- e = leaky_relu(as[src]+ad[dst]); m = segmax
__global__ void k_edge_logits_max(const float* __restrict__ as_, const float* __restrict__ ad_,
                                  const int* __restrict__ src, const int* __restrict__ dst,
                                  float* __restrict__ e, float* __restrict__ m,
                                  int nE, int H) {
  int i = blockIdx.x * blockDim.x + threadIdx.x;
  if (i >= nE * H) return;
  int eidx = i / H, h = i - eidx * H;
  float v = as_[src[eidx] * H + h] + ad_[dst[eidx] * H + h];
  v = v > 0.f ? v : 0.2f * v;
  e[i] = v;
  atomicMaxF(&m[dst[eidx] * H + h], v);
}

// ---------------------------------- ex = exp(e - m[dst]); s = segsum(ex)
__global__ void k_edge_exp_sum(const int* __restrict__ dst, float* __restrict__ e,
                               const float* __restrict__ m, float* __restrict__ s,
                               int nE, int H) {
  int i = blockIdx.x * blockDim.x + threadIdx.x;
  if (i >= nE * H) return;
  int eidx = i / H, h = i - eidx * H;
  int di = dst[eidx] * H + h;
  float ex = expf(e[i] - m[di]);
  e[i] = ex;
  atomicAdd(&s[di], ex);
}

// --------------------- out[dst] += hs[src] * (ex / (s[dst]+1e-16)), x4
__global__ void k_edge_aggregate(const int* __restrict__ src, const int* __restrict__ dst,
                                 const float* __restrict__ ex, const float* __restrict__ s,
                                 const float* __restrict__ hs, float* __restrict__ out,
                                 int nE, int D, int C, int H) {
  int i = blockIdx.x * blockDim.x + threadIdx.x;
  int q = D >> 2;
  if (i >= nE * q) return;
  int eidx = i / q;
  int f = (i - eidx * q) << 2;
  int h = f / C;
  int d = dst[eidx];
  float alpha = ex[(size_t)eidx * H + h] / (s[(size_t)d * H + h] + 1e-16f);
  const float4 hv = *(const float4*)(hs + (size_t)src[eidx] * D + f);  // b128
  float* op = out + (size_t)d * D + f;
  atomicAdd(op + 0, hv.x * alpha);
  atomicAdd(op + 1, hv.y * alpha);
  atomicAdd(op + 2, hv.z * alpha);
  atomicAdd(op + 3, hv.w * alpha);
}

// -------------------------------------------------- out += b ; opt relu
__global__ void k_bias_act(float* __restrict__ out, const float* __restrict__ b,
                           int n, int D, int relu) {
  int i = blockIdx.x * blockDim.x + threadIdx.x;
  if (i >= n * D) return;
  float v = out[i] + b[i % D];
  if (relu) v = v > 0.f ? v : 0.f;
  out[i] = v;
}

// ------------- decoder GEMM: relu(concat(zu2[lu],zm2[lm]) @ dw1 + db1)
// Gather is fused into the A-operand loads. K=1024, N=16 fixed.
__global__ void k_dec_gemm(const float* __restrict__ ZU2, const float* __restrict__ ZM2,
                           const int* __restrict__ labu, const int* __restrict__ labm,
                           const float* __restrict__ W, const float* __restrict__ bias,
                           float* __restrict__ Hout, int M) {
  const int K = 1024, N = 16, HD = 512;
  int wave = blockIdx.x * (blockDim.x >> 5) + (threadIdx.x >> 5);
  int tiles = M >> 4;  // tilesN == 1
  if (wave >= tiles) return;
  int mt = wave;
  int lane = threadIdx.x & 31;
  int hf = lane >> 4, l15 = lane & 15;
  int row = mt * 16 + l15;
  size_t bu = (size_t)labu[row] * HD;
  size_t bm = (size_t)labm[row] * HD;
  int ncol = l15;
  v8f acc = {};
  for (int k = 0; k < K; k += 4) {
    int kb = k + hf * 2;  // kb even -> kb,kb+1 never straddle the 512 boundary
    v2f a, b;
    const float* s0 = (kb < HD) ? (ZU2 + bu + kb) : (ZM2 + bm + (kb - HD));
    a[0] = s0[0];
    a[1] = s0[1];
    b[0] = W[(size_t)kb * N + ncol];
    b[1] = W[(size_t)(kb + 1) * N + ncol];
    acc = __builtin_amdgcn_wmma_f32_16x16x4_f32(false, a, false, b, (short)0, acc,
                                                false, false);
  }
  float bv = bias[ncol];
  int row0 = mt * 16 + hf * 8;
#pragma unroll
  for (int r = 0; r < 8; ++r) {
    float v = acc[r] + bv;
    v = v > 0.f ? v : 0.f;
    Hout[(size_t)(row0 + r) * N + ncol] = v;
  }
}

// ------------------------------------------ final: out = H @ dw2 + db2
__global__ void k_dec_out(const float* __restrict__ Hh, const float* __restrict__ dw2,
                          const float* __restrict__ db2, float* __restrict__ out, int M) {
  int i = blockIdx.x * blockDim.x + threadIdx.x;
  if (i >= M) return;
  float acc = db2[0];
  const float* hp = Hh + (size_t)i * 16;
#pragma unroll
  for (int j = 0; j < 16; ++j) acc += hp[j] * dw2[j];
  out[i] = acc;
}

// =================================================== host orchestration
static void launch_gemm(const float* A, const float* B, const float* bias, float* C,
                        int M, int K, int N, int act, hipStream_t st) {
  if ((N & 63) == 0 && (K & (KT - 1)) == 0) {
    int blocks = (M / 16) * (N / 64);
    k_gemm_f32_lds<<<blocks, 128, 0, st>>>(A, B, bias, C, M, K, N, act);
  } else {
    int tiles = (M / 16) * (N / 16);
    k_gemm_f32<<<ceil_div_i(tiles, 4), 128, 0, st>>>(A, B, bias, C, M, K, N, act);
  }
}

static void launch_fill(float* p, float v, int n, hipStream_t st) {
  k_fill<<<ceil_div_i(n, 256), 256, 0, st>>>(p, v, n);
}

struct GatScratch { float *hs, *hd, *as_, *ad_, *e, *m, *s; };

static void run_gat(const float* xs, int Ns, int Fs,
                    const float* xd, int Nd, int Fd,
                    const int* src, const int* dst, int nE,
                    const float* Ws, const float* Wd,
                    const float* a_s, const float* a_d, const float* b,
                    int H, int C, float* out, int relu,
                    const GatScratch& t, hipStream_t st) {
  int D = H * C;
  launch_gemm(xs, Ws, nullptr, t.hs, Ns, Fs, D, 0, st);
  launch_gemm(xd, Wd, nullptr, t.hd, Nd, Fd, D, 0, st);
  k_rowdot<<<ceil_div_i(Ns * H, 256), 256, 0, st>>>(t.hs, a_s, t.as_, Ns, H, C);
  k_rowdot<<<ceil_div_i(Nd * H, 256), 256, 0, st>>>(t.hd, a_d, t.ad_, Nd, H, C);
  launch_fill(t.m, -INFINITY, Nd * H, st);
  launch_fill(t.s, 0.f, Nd * H, st);
  launch_fill(out, 0.f, Nd * D, st);
  k_edge_logits_max<<<ceil_div_i(nE * H, 256), 256, 0, st>>>(t.as_, t.ad_, src, dst,
                                                             t.e, t.m, nE, H);
  k_edge_exp_sum<<<ceil_div_i(nE * H, 256), 256, 0, st>>>(dst, t.e, t.m, t.s, nE, H);
  int total = nE * (D / 4);
  k_edge_aggregate<<<ceil_div_i(total, 256), 256, 0, st>>>(src, dst, t.e, t.s, t.hs,
                                                           out, nE, D, C, H);
  k_bias_act<<<ceil_div_i(Nd * D, 256), 256, 0, st>>>(out, b, Nd, D, relu);
}

extern "C" void kernel_launch(void* const* d_in, const int* in_sizes, int n_in,
                              void* d_out, int out_size, void* d_ws, size_t ws_size,
                              hipStream_t stream) {
  (void)in_sizes; (void)n_in; (void)out_size; (void)ws_size;
  const float* xu = (const float*)d_in[0];
  const float* xm = (const float*)d_in[1];
  const int* um_src = (const int*)d_in[2];
  const int* um_dst = (const int*)d_in[3];
  const int* mu_src = (const int*)d_in[4];
  const int* mu_dst = (const int*)d_in[5];
  const int* lab_u  = (const int*)d_in[6];
  const int* lab_m  = (const int*)d_in[7];
  const float *w1um_s = (const float*)d_in[8],  *w1um_d = (const float*)d_in[9];
  const float *a1um_s = (const float*)d_in[10], *a1um_d = (const float*)d_in[11];
  const float *b1um   = (const float*)d_in[12];
  const float *w1mu_s = (const float*)d_in[13], *w1mu_d = (const float*)d_in[14];
  const float *a1mu_s = (const float*)d_in[15], *a1mu_d = (const float*)d_in[16];
  const float *b1mu   = (const float*)d_in[17];
  const float *w2um_s = (const float*)d_in[18], *w2um_d = (const float*)d_in[19];
  const float *a2um_s = (const float*)d_in[20], *a2um_d = (const float*)d_in[21];
  const float *b2um   = (const float*)d_in[22];
  const float *w2mu_s = (const float*)d_in[23], *w2mu_d = (const float*)d_in[24];
  const float *a2mu_s = (const float*)d_in[25], *a2mu_d = (const float*)d_in[26];
  const float *b2mu   = (const float*)d_in[27];
  const float *dw1 = (const float*)d_in[28], *db1 = (const float*)d_in[29];
  const float *dw2 = (const float*)d_in[30], *db2 = (const float*)d_in[31];
  float* out = (float*)d_out;

  // ---- workspace carve (peak ~750 MB, scratch reused across layers) ----
  char* w = (char*)d_ws;
  auto take = [&](size_t bytes) -> float* {
    float* p = (float*)w;
    w += (bytes + 255) & ~(size_t)255;
    return p;
  };
  float* S_HS = take((size_t)NU_N * 512 * 4);  // src projections (max layer2)
  float* S_HD = take((size_t)NU_N * 512 * 4);  // dst projections (max layer2)
  float* ZU   = take((size_t)NU_N * 128 * 4);
  float* ZM   = take((size_t)NM_N * 128 * 4);
  float* ZU2  = take((size_t)NU_N * 512 * 4);
  float* ZM2  = take((size_t)NM_N * 512 * 4);
  float* S_AS = take((size_t)NU_N * 8 * 4);
  float* S_AD = take((size_t)NU_N * 8 * 4);
  float* S_M  = take((size_t)NU_N * 8 * 4);
  float* S_S  = take((size_t)NU_N * 8 * 4);
  float* S_E  = take((size_t)E_N * 8 * 4);
  float* DECH = take((size_t)EL_N * 16 * 4);

  GatScratch t{S_HS, S_HD, S_AS, S_AD, S_E, S_M, S_S};

  // layer 1 (relu outside bias-add)
  run_gat(xu, NU_N, 32,  xm, NM_N, 128, um_src, um_dst, E_N,
          w1um_s, w1um_d, a1um_s, a1um_d, b1um, 8, 16, ZM, 1, t, stream);
  run_gat(xm, NM_N, 128, xu, NU_N, 32,  mu_src, mu_dst, E_N,
          w1mu_s, w1mu_d, a1mu_s, a1mu_d, b1mu, 8, 16, ZU, 1, t, stream);
  // layer 2 (no relu)
  run_gat(ZU, NU_N, 128, ZM, NM_N, 128, um_src, um_dst, E_N,
          w2um_s, w2um_d, a2um_s, a2um_d, b2um, 4, 128, ZM2, 0, t, stream);
  run_gat(ZM, NM_N, 128, ZU, NU_N, 128, mu_src, mu_dst, E_N,
          w2mu_s, w2mu_d, a2mu_s, a2mu_d, b2mu, 4, 128, ZU2, 0, t, stream);
  // decoder
  k_dec_gemm<<<ceil_div_i(EL_N / 16, 4), 128, 0, stream>>>(ZU2, ZM2, lab_u, lab_m,
                                                           dw1, db1, DECH, EL_N);
  k_dec_out<<<ceil_div_i(EL_N, 256), 256, 0, stream>>>(DECH, dw2, db2, out, EL_N);
}